// HomoGNNModel_51908974739743
// MI455X (gfx1250) — compile-verified
//
#include <hip/hip_runtime.h>

typedef __attribute__((ext_vector_type(2))) float v2f;
typedef __attribute__((ext_vector_type(8))) float v8f;

constexpr int N0   = 292864;
constexpr int N1   = 11264;
constexpr int N2   = 1024;
constexpr int F0   = 25;
constexpr int F1   = 10;
constexpr int DIN  = 256;   // embedding dim
constexpr int HID  = 256;
constexpr int NCLS = 47;

// ---------------------------------------------------------------------------
// Kernel A: fused embedding-gather + neighbor-mean for layer 0.
// Builds xcat[j] = [ emb[gids0[j]] (256) | mean_k emb[gids0[nbr0[j,k]]] (256) ]
// One block (64 threads, 2 waves) per destination node; float4 row traffic.
// ---------------------------------------------------------------------------
__global__ void gather_agg0(const int* __restrict__ gids0,
                            const int* __restrict__ nbr0,
                            const float* __restrict__ emb,
                            float* __restrict__ xcat) {
    const int j = blockIdx.x;
    const int t = threadIdx.x;              // 0..63, one float4 per thread
    __shared__ int sgid[F0];
    __shared__ int sdst;
    if (t < F0) sgid[t] = gids0[nbr0[j * F0 + t]];
    if (t == 0) sdst = gids0[j];
    __syncthreads();

    const float4* embv = (const float4*)emb;     // row = 64 float4
    float4* xv = (float4*)xcat;                  // row = 128 float4 (K=512)

    float4 self = embv[(size_t)sdst * (DIN / 4) + t];

    float sx = 0.f, sy = 0.f, sz = 0.f, sw = 0.f;
    #pragma unroll
    for (int k = 0; k < F0; ++k) {
        float4 e = embv[(size_t)sgid[k] * (DIN / 4) + t];
        sx += e.x; sy += e.y; sz += e.z; sw += e.w;
    }
    const float inv = 1.0f / (float)F0;
    float4 agg; agg.x = sx * inv; agg.y = sy * inv; agg.z = sz * inv; agg.w = sw * inv;

    xv[(size_t)j * 128 + t]      = self;
    xv[(size_t)j * 128 + 64 + t] = agg;
}

// ---------------------------------------------------------------------------
// Kernel C: layer-1 concat builder from h (no embedding indirection).
// xcat[j] = [ h[j] (256) | mean_k h[nbr1[j,k]] (256) ]
// ---------------------------------------------------------------------------
__global__ void gather_agg1(const int* __restrict__ nbr1,
                            const float* __restrict__ h,
                            float* __restrict__ xcat) {
    const int j = blockIdx.x;
    const int t = threadIdx.x;              // 0..63
    __shared__ int snb[F1];
    if (t < F1) snb[t] = nbr1[j * F1 + t];
    __syncthreads();

    const float4* hv = (const float4*)h;
    float4* xv = (float4*)xcat;

    float4 self = hv[(size_t)j * (HID / 4) + t];

    float sx = 0.f, sy = 0.f, sz = 0.f, sw = 0.f;
    #pragma unroll
    for (int k = 0; k < F1; ++k) {
        float4 e = hv[(size_t)snb[k] * (HID / 4) + t];
        sx += e.x; sy += e.y; sz += e.z; sw += e.w;
    }
    const float inv = 1.0f / (float)F1;
    float4 agg; agg.x = sx * inv; agg.y = sy * inv; agg.z = sz * inv; agg.w = sw * inv;

    xv[(size_t)j * 128 + t]      = self;
    xv[(size_t)j * 128 + 64 + t] = agg;
}

// ---------------------------------------------------------------------------
// SAGE GEMM tile kernel via V_WMMA_F32_16X16X4_F32, templated on NC so every
// stride is an immediate constant (folds into global_load IOFFSET).
//   out[M x NC] = relu?( xcat[M x 512] * [Wself ; Wneigh] + bias )
// One wave per 16x16 output tile; 128 x (16x16x4) WMMAs per tile.
//
// Per ISA 7.12.2 (wave32, f32):
//   A 16x4 : lane holds M = lane&15 ; float2 = A[m][k + 2*(lane>=16) .. +1]
//   B 4x16 : lane holds N = lane&15 ; b.x/b.y = B[k + 2*(lane>=16) (+1)][n]
//   C 16x16: VGPR i -> row (i + 8*(lane>=16)), col (lane&15)
//
// Edge tile (NC=47): column index is CLAMPED, not masked — out-of-range lanes
// compute garbage columns that are simply never stored, so all loads and the
// WMMA run at full EXEC (required by ISA 7.12) with zero divergence.
// ---------------------------------------------------------------------------
template<int NC, bool RELU>
__global__ void sage_gemm(const float* __restrict__ xcat,   // M x 512
                          const float* __restrict__ Wself,  // 256 x NC
                          const float* __restrict__ Wneigh, // 256 x NC
                          const float* __restrict__ bias,   // NC
                          float* __restrict__ out) {        // M x NC
    const int lane  = threadIdx.x & 31;
    const int half  = lane >> 4;        // 0: lanes 0-15, 1: lanes 16-31
    const int l16   = lane & 15;
    const int tileM = blockIdx.x * 16;
    const int n     = blockIdx.y * 16 + l16;
    const int ncl   = (n < NC) ? n : (NC - 1);   // clamped column (see note)

    v8f c = {};
    const float* ap  = xcat   + (size_t)(tileM + l16) * 512 + 2 * half;
    const float* wp0 = Wself  + 2 * half * NC + ncl;
    const float* wp1 = Wneigh + 2 * half * NC + ncl;

    #pragma unroll 8
    for (int k = 0; k < 256; k += 4) {
        v2f a = *(const v2f*)(ap + k);          // A[m][k+2h .. k+2h+1]
        v2f b;
        b.x = wp0[(size_t)k * NC];              // Wself[k+2h  ][n]
        b.y = wp0[(size_t)k * NC + NC];         // Wself[k+2h+1][n]
        c = __builtin_amdgcn_wmma_f32_16x16x4_f32(
                false, a, false, b, (short)0, c, false, false);
    }
    #pragma unroll 8
    for (int k = 0; k < 256; k += 4) {
        v2f a = *(const v2f*)(ap + 256 + k);    // agg half of xcat
        v2f b;
        b.x = wp1[(size_t)k * NC];              // Wneigh[k+2h  ][n]
        b.y = wp1[(size_t)k * NC + NC];         // Wneigh[k+2h+1][n]
        c = __builtin_amdgcn_wmma_f32_16x16x4_f32(
                false, a, false, b, (short)0, c, false, false);
    }

    if (n < NC) {
        const float bi = bias[n];
        #pragma unroll
        for (int i = 0; i < 8; ++i) {
            const int row = tileM + i + 8 * half;
            float v = c[i] + bi;
            if (RELU) v = fmaxf(v, 0.0f);
            out[(size_t)row * NC + n] = v;
        }
    }
}

// ---------------------------------------------------------------------------
// Launch: gather+agg0 -> GEMM0(relu) -> gather+agg1 -> GEMM1
// ---------------------------------------------------------------------------
extern "C" void kernel_launch(void* const* d_in, const int* in_sizes, int n_in,
                              void* d_out, int out_size, void* d_ws, size_t ws_size,
                              hipStream_t stream) {
    const int*   gids0   = (const int*)d_in[0];
    const int*   nbr0    = (const int*)d_in[1];
    const int*   nbr1    = (const int*)d_in[2];
    const float* emb     = (const float*)d_in[3];
    const float* Wself0  = (const float*)d_in[4];
    const float* Wneigh0 = (const float*)d_in[5];
    const float* b0      = (const float*)d_in[6];
    const float* Wself1  = (const float*)d_in[7];
    const float* Wneigh1 = (const float*)d_in[8];
    const float* b1      = (const float*)d_in[9];
    float* out = (float*)d_out;

    // Workspace layout (reused): h first, then xcat (layer-1 xcat reuses layer-0's slot)
    float* h    = (float*)d_ws;                 // N1 * 256 floats (11.5 MB)
    float* xcat = h + (size_t)N1 * HID;         // N1 * 512 floats (23 MB)

    // Layer 0: fused gather + mean over 25 neighbors (skips materializing x entirely)
    gather_agg0<<<N1, 64, 0, stream>>>(gids0, nbr0, emb, xcat);
    // h = relu(xcat * [Wself0; Wneigh0] + b0)
    sage_gemm<HID, true><<<dim3(N1 / 16, HID / 16), 32, 0, stream>>>(xcat, Wself0, Wneigh0, b0, h);

    // Layer 1
    gather_agg1<<<N2, 64, 0, stream>>>(nbr1, h, xcat);
    sage_gemm<NCLS, false><<<dim3(N2 / 16, (NCLS + 15) / 16), 32, 0, stream>>>(xcat, Wself1, Wneigh1, b1, out);
}